// _WeightBlock_3083786518782
// MI455X (gfx1250) — compile-verified
//
#include <hip/hip_runtime.h>
#include <math.h>

// ---------------------------------------------------------------------------
// Problem constants (match reference)
// ---------------------------------------------------------------------------
#define NNODES 50000
#define NEDGES 800000
#define DIM    128            // IN_DIM == OUT_DIM == 128

typedef __attribute__((ext_vector_type(2))) float v2f;
typedef __attribute__((ext_vector_type(8))) float v8f;

// ---------------------------------------------------------------------------
// Kernel 0: zero the accumulation workspace (sums [N*128] + cnt [N])
// ---------------------------------------------------------------------------
__global__ __launch_bounds__(256)
void gnn_zero_ws(float* __restrict__ sums_and_cnt, int n_elems) {
    int i = blockIdx.x * blockDim.x + threadIdx.x;
    if (i < n_elems) sums_and_cnt[i] = 0.0f;
}

// ---------------------------------------------------------------------------
// Kernel 1: fused dual GEMM via WMMA (fp32, 16x16x4)
//   G[m,n] = sum_k x[m,k] * Wg[n,k]
//   C[m,n] = sum_k x[m,k] * (Wg[n,k] + Wl[n,k])
// One block = one 16-row M tile (grid.x = 50000/16 = 3125).
// 8 waves per block, wave w owns N-tile w (cols 16w..16w+15).
// x tile staged in LDS once per block (read by all 8 waves).
// ---------------------------------------------------------------------------
#define XS_STRIDE 132   // 16 rows * 132 floats; bank-conflict-free A reads

__global__ __launch_bounds__(256)
void gnn_dual_gemm_wmma(const float* __restrict__ x,
                        const float* __restrict__ Wg,
                        const float* __restrict__ Wl,
                        float* __restrict__ G,
                        float* __restrict__ C) {
    __shared__ float xs[16 * XS_STRIDE];

    const int tid  = threadIdx.x;
    const int m0   = blockIdx.x * 16;          // first row of this M tile

    // --- cooperative load of x tile (16 x 128 f32 = 8 KB) into LDS ---
    // 2048 floats = 512 float4; 256 threads x 2 float4 each, coalesced.
    {
        const float* src = x + (size_t)m0 * DIM;
        #pragma unroll
        for (int t = 0; t < 2; ++t) {
            int q   = tid + t * 256;           // float4 index 0..511
            int row = (q * 4) / DIM;
            int col = (q * 4) % DIM;
            const float4 v = *reinterpret_cast<const float4*>(src + row * DIM + col);
            float* d = &xs[row * XS_STRIDE + col];
            d[0] = v.x; d[1] = v.y; d[2] = v.z; d[3] = v.w;
        }
    }
    __syncthreads();

    const int lane = tid & 31;
    const int wv   = tid >> 5;                 // wave id 0..7 -> N tile
    const int half = lane >> 4;                // 0: K pair {0,1}, 1: K pair {2,3}
    const int r    = lane & 15;                // A: row within tile / B: col within tile
    const int n    = wv * 16 + r;              // global output column for B fragment

    v8f accG = {0.f, 0.f, 0.f, 0.f, 0.f, 0.f, 0.f, 0.f};
    v8f accC = {0.f, 0.f, 0.f, 0.f, 0.f, 0.f, 0.f, 0.f};

    const float* wg_row = Wg + (size_t)n * DIM;
    const float* wl_row = Wl + (size_t)n * DIM;

    #pragma unroll 8
    for (int k0 = 0; k0 < DIM; k0 += 4) {
        const int ka = k0 + 2 * half;
        // A fragment: x[m0 + r, ka..ka+1]  (from LDS)
        v2f a = *reinterpret_cast<const v2f*>(&xs[r * XS_STRIDE + ka]);
        // B fragments: B[k, n] = W[n, k] -> row-major read of W row n
        v2f bg = *reinterpret_cast<const v2f*>(wg_row + ka);
        v2f bl = *reinterpret_cast<const v2f*>(wl_row + ka);
        v2f bc = bg + bl;
        accG = __builtin_amdgcn_wmma_f32_16x16x4_f32(
            false, a, false, bg, (short)0, accG, false, false);
        accC = __builtin_amdgcn_wmma_f32_16x16x4_f32(
            false, a, false, bc, (short)0, accC, false, false);
    }

    // D layout: VGPR j -> row (j + 8*half), col r  within the 16x16 tile
    const int col = wv * 16 + r;
    #pragma unroll
    for (int j = 0; j < 8; ++j) {
        const int row = m0 + j + 8 * half;
        G[(size_t)row * DIM + col] = accG[j];
        C[(size_t)row * DIM + col] = accC[j];
    }
}

// ---------------------------------------------------------------------------
// Kernel 2: edge gather + sorted-segment sum with run-compressed atomics.
// One wave processes a contiguous span of edges; lane l owns output
// channels [4l, 4l+4). Runs of equal seg_id are accumulated in registers
// and flushed once with global_atomic_add_f32.
// ---------------------------------------------------------------------------
#define SCAT_BLOCKS 500
#define SCAT_WAVES  (SCAT_BLOCKS * 8)          // 4000 waves
#define SCAT_SPAN   (NEDGES / SCAT_WAVES)      // 200 edges / wave (exact)

__global__ __launch_bounds__(256)
void gnn_edge_segsum(const float* __restrict__ comb,        // C table [N,128]
                     const int*   __restrict__ neigh_idx,
                     const int*   __restrict__ seg_ids,
                     float*       __restrict__ sums,
                     float*       __restrict__ cnt) {
    const int lane = threadIdx.x & 31;
    int wflat = (int)(blockIdx.x * 8) + (int)(threadIdx.x >> 5);
    wflat = __builtin_amdgcn_readfirstlane(wflat);   // wave-uniform -> s_loads

    const int e0 = wflat * SCAT_SPAN;
    const int e1 = (e0 + SCAT_SPAN < NEDGES) ? (e0 + SCAT_SPAN) : NEDGES;
    const int c  = lane * 4;                         // channel base for this lane

    float a0 = 0.f, a1 = 0.f, a2 = 0.f, a3 = 0.f;
    float runc = 0.f;
    int   cur  = -1;

    for (int e = e0; e < e1; ++e) {
        const int seg = seg_ids[e];                  // uniform scalar load
        const int idx = neigh_idx[e];                // uniform scalar load
        if (seg != cur) {                            // uniform branch
            if (cur >= 0) {
                float* dst = sums + (size_t)cur * DIM + c;
                atomicAdd(dst + 0, a0);
                atomicAdd(dst + 1, a1);
                atomicAdd(dst + 2, a2);
                atomicAdd(dst + 3, a3);
                if (lane == 0) atomicAdd(cnt + cur, runc);
            }
            cur = seg; a0 = a1 = a2 = a3 = 0.f; runc = 0.f;
        }
        const float4 v = *reinterpret_cast<const float4*>(
            comb + (size_t)idx * DIM + c);           // coalesced 512B row gather
        a0 += v.x; a1 += v.y; a2 += v.z; a3 += v.w;
        runc += 1.0f;
    }
    if (cur >= 0) {
        float* dst = sums + (size_t)cur * DIM + c;
        atomicAdd(dst + 0, a0);
        atomicAdd(dst + 1, a1);
        atomicAdd(dst + 2, a2);
        atomicAdd(dst + 3, a3);
        if (lane == 0) atomicAdd(cnt + cur, runc);
    }
}

// ---------------------------------------------------------------------------
// Kernel 3: finalize.  out = elu( (cnt>0 ? sums/cnt : G) + G + bias )
// ---------------------------------------------------------------------------
__global__ __launch_bounds__(256)
void gnn_finalize(const float* __restrict__ sums,
                  const float* __restrict__ cnt,
                  const float* __restrict__ G,
                  const float* __restrict__ bias,
                  float* __restrict__ out) {
    int i = blockIdx.x * blockDim.x + threadIdx.x;
    if (i >= NNODES * DIM) return;
    const int node = i >> 7;
    const int col  = i & (DIM - 1);
    const float g  = G[i];
    const float cn = cnt[node];
    const float m  = (cn > 0.f) ? (sums[i] / cn) : g;
    const float v  = m + g + bias[col];
    out[i] = (v > 0.f) ? v : expm1f(v);
}

// ---------------------------------------------------------------------------
// Host launcher
// ---------------------------------------------------------------------------
extern "C" void kernel_launch(void* const* d_in, const int* in_sizes, int n_in,
                              void* d_out, int out_size, void* d_ws, size_t ws_size,
                              hipStream_t stream) {
    const float* x         = (const float*)d_in[0];
    const float* Wg        = (const float*)d_in[1];
    const float* Wl        = (const float*)d_in[2];
    const float* bias      = (const float*)d_in[3];
    const int*   neigh_idx = (const int*)d_in[4];
    const int*   seg_ids   = (const int*)d_in[5];
    float*       out       = (float*)d_out;

    // Workspace layout (floats): G | C | sums | cnt
    float* ws   = (float*)d_ws;
    const size_t NM = (size_t)NNODES * DIM;
    float* G    = ws;
    float* C    = ws + NM;
    float* sums = ws + 2 * NM;
    float* cnt  = ws + 3 * NM;

    // 0) zero sums + cnt (poisoned workspace; must re-zero every call)
    const int zero_elems = (int)(NM + NNODES);
    gnn_zero_ws<<<(zero_elems + 255) / 256, 256, 0, stream>>>(sums, zero_elems);

    // 1) fused dual GEMM: G = x@Wg.T, C = x@(Wg+Wl).T   (WMMA fp32)
    gnn_dual_gemm_wmma<<<NNODES / 16, 256, 0, stream>>>(x, Wg, Wl, G, C);

    // 2) edge gather + sorted-segment sum
    gnn_edge_segsum<<<SCAT_BLOCKS, 256, 0, stream>>>(C, neigh_idx, seg_ids, sums, cnt);

    // 3) finalize with ELU
    gnn_finalize<<<((int)NM + 255) / 256, 256, 0, stream>>>(sums, cnt, G, bias, out);
}